// LSTM_price_45896020525788
// MI455X (gfx1250) — compile-verified
//
#include <hip/hip_runtime.h>
#include <math.h>

// ---------------------------------------------------------------------------
// DA-RNN style LSTM (T=10, D=H=45) fully fused into ONE persistent workgroup.
//  - all encoder weights staged once into LDS (dynamic ~80KB; WGP has 320KB)
//  - decoder-phase global weights prefetched (global_prefetch_b8) while the
//    encoder recurrence runs
//  - encoder scan runs on-chip with workgroup barriers between steps
//  - 180x90 gate GEMV per step on the matrix pipe: V_WMMA_F32_16X16X4_F32
//    (full f32 precision, N=1 in column 0); operands fetched as aligned
//    ds_load_b64 in all lanes, lane-masked with v_cndmask (no EXEC juggling)
//  - all small reductions/softmaxes are wave32 shfl_xor butterflies
// ---------------------------------------------------------------------------

typedef __attribute__((ext_vector_type(2))) float v2f;
typedef __attribute__((ext_vector_type(8))) float v8f;

#define TT 10
#define DD 45
#define HH 45
#define NG 180   // 4*H gate rows
#define KV 90    // 2*H (wi | h)
#define KP 92    // K padded to multiple of 4
#define MP 192   // M padded to multiple of 16
#define NTHREADS 256

__device__ __forceinline__ float sigmoidf_(float x) {
    return 1.0f / (1.0f + __expf(-x));
}

__device__ __forceinline__ float waveSum(float x) {
#pragma unroll
    for (int off = 16; off > 0; off >>= 1) x += __shfl_xor(x, off, 32);
    return x;
}

__device__ __forceinline__ float waveMax(float x) {
#pragma unroll
    for (int off = 16; off > 0; off >>= 1) x = fmaxf(x, __shfl_xor(x, off, 32));
    return x;
}

__global__ void __launch_bounds__(NTHREADS)
lstm_price_fused(const float* __restrict__ price,   // [T,D]
                 const float* __restrict__ Wa,      // [1,2H+T]
                 const float* __restrict__ ba,      // [1]
                 const float* __restrict__ Wih_e,   // [4H,D]
                 const float* __restrict__ Whh_e,   // [4H,H]
                 const float* __restrict__ bih_e,   // [4H]
                 const float* __restrict__ bhh_e,   // [4H]
                 const float* __restrict__ W1,      // [D,2H+D] (=45x135)
                 const float* __restrict__ b1,      // [D]
                 const float* __restrict__ W2,      // [1,D]
                 const float* __restrict__ b2,      // [1]
                 const float* __restrict__ Wih_d,   // [4H,D]
                 const float* __restrict__ bih_d,   // [4H]
                 const float* __restrict__ bhh_d,   // [4H]
                 const float* __restrict__ Wfc,     // [45,2H]
                 const float* __restrict__ bfc,     // [45]
                 float* __restrict__ out)           // [45]
{
    extern __shared__ float smem[];
    float* Wg     = smem;                 // [MP][KP] combined [Wih_e|Whh_e], zero padded
    float* xL     = Wg     + MP * KP;     // [T][D]
    float* hsb    = xL     + TT * DD;     // [T][H] encoder hidden states
    float* be     = hsb    + TT * HH;     // [NG]  bih_e + bhh_e
    float* v      = be     + NG;          // [KP]  (wi | h | pad)
    float* gates  = v      + KP;          // [MP]
    float* cvec   = gates  + MP;          // [H]
    float* scores = cvec   + HH;          // [D]
    float* red    = scores + DD;          // [64] reduction scratch
    float* dtmp   = red    + 64;          // [T][D] decoder attn temp
    float* dscore = dtmp   + TT * DD;     // [16]
    float* ctx    = dscore + 16;          // [H]
    float* dg     = ctx    + HH;          // [NG]
    float* h1v    = dg     + NG;          // [H]

    const int tid  = threadIdx.x;
    const int wave = tid >> 5;
    const int lane = tid & 31;
    const int lhalf = lane >> 4;   // 0: lanes 0-15 (K,K+1) ; 1: lanes 16-31 (K+2,K+3)
    const int lrow  = lane & 15;   // A: M row ; B/D: N column
    const bool col0 = (lrow == 0);

    // ---- warm L2/WGP$ for the decoder phase while encoder runs -----------
    if (wave == 7) {
        for (int i = lane * 16; i < DD * (2 * HH + DD); i += 32 * 16)
            __builtin_prefetch(&W1[i], 0, 1);
        for (int i = lane * 16; i < NG * DD; i += 32 * 16)
            __builtin_prefetch(&Wih_d[i], 0, 1);
        for (int i = lane * 16; i < 45 * 2 * HH; i += 32 * 16)
            __builtin_prefetch(&Wfc[i], 0, 1);
    }

    // ---- stage weights / inputs into LDS --------------------------------
    for (int i = tid; i < MP * KP; i += NTHREADS) Wg[i] = 0.0f;
    __syncthreads();
    for (int i = tid; i < NG * KV; i += NTHREADS) {
        int g = i / KV, k = i % KV;
        Wg[g * KP + k] = (k < HH) ? Wih_e[g * DD + k] : Whh_e[g * HH + (k - HH)];
    }
    for (int i = tid; i < TT * DD; i += NTHREADS) xL[i] = price[i];
    if (tid < NG) be[tid] = bih_e[tid] + bhh_e[tid];
    if (tid < HH) cvec[tid] = 0.0f;
    if (tid < 47) v[HH + tid] = 0.0f;   // h = 0 and K-pad = 0
    __syncthreads();

    // ---- encoder scan ----------------------------------------------------
    for (int t = 0; t < TT; ++t) {
        // shared scalar: ba + Wa[0:45].h + Wa[45:90].c  (same for every d)
        if (wave == 0) {
            float p = Wa[lane] * v[HH + lane] + Wa[HH + lane] * cvec[lane];
            if (lane + 32 < HH)
                p += Wa[lane + 32] * v[HH + lane + 32] + Wa[HH + lane + 32] * cvec[lane + 32];
            else if (lane >= HH)   // lanes 45..31? (none) -- lanes 0..31 all < 45
                p = 0.0f;
            p = waveSum(p);
            if (lane == 0) red[63] = p + ba[0];
        }
        __syncthreads();
        // scores[d] = base + sum_s Wa[90+s] * x[s,d]
        if (tid < DD) {
            float s = red[63];
#pragma unroll
            for (int ss = 0; ss < TT; ++ss) s += Wa[2 * HH + ss] * xL[ss * DD + tid];
            scores[tid] = s;
        }
        __syncthreads();
        // softmax over D=45, wave 0 butterfly
        if (wave == 0) {
            float s0 = scores[lane];                                     // lanes 0..31
            float s1 = (lane + 32 < DD) ? scores[lane + 32] : -1e30f;
            float m  = waveMax(fmaxf(s0, s1));
            float e0 = __expf(s0 - m);
            float e1 = (lane + 32 < DD) ? __expf(s1 - m) : 0.0f;
            float inv = 1.0f / waveSum(e0 + e1);
            scores[lane] = e0 * inv;
            if (lane + 32 < DD) scores[lane + 32] = e1 * inv;
        }
        __syncthreads();
        if (tid < DD) v[tid] = scores[tid] * xL[t * DD + tid];   // wi
        __syncthreads();

        // gates[0:180] = Wg(192x92) * v(92)  via V_WMMA_F32_16X16X4_F32, N=1
        for (int tile = wave; tile < MP / 16; tile += NTHREADS / 32) {
            v8f acc = {};
            const float* wrow = &Wg[(tile * 16 + lrow) * KP];
#pragma unroll
            for (int kc = 0; kc < KP / 4; ++kc) {
                const int kb = kc * 4 + lhalf * 2;
                const v2f a  = *(const v2f*)&wrow[kb];   // aligned ds_load_b64
                const v2f bv = *(const v2f*)&v[kb];      // aligned ds_load_b64 (all lanes)
                v2f b;
                b.x = col0 ? bv.x : 0.0f;                // v_cndmask, EXEC untouched
                b.y = col0 ? bv.y : 0.0f;
                acc = __builtin_amdgcn_wmma_f32_16x16x4_f32(
                        false, a, false, b, (short)0, acc, false, false);
            }
            // D: VGPR r -> row r (lanes 0-15) / row r+8 (lanes 16-31); col 0 = lrow==0
            if (col0) {
                const int base = tile * 16 + lhalf * 8;
#pragma unroll
                for (int r = 0; r < 8; ++r) gates[base + r] = acc[r];
            }
        }
        __syncthreads();

        // element-wise LSTM update (gate order i,f,g,o)
        if (tid < HH) {
            float gi = gates[tid]            + be[tid];
            float gf = gates[HH + tid]       + be[HH + tid];
            float gg = gates[2 * HH + tid]   + be[2 * HH + tid];
            float go = gates[3 * HH + tid]   + be[3 * HH + tid];
            float c  = sigmoidf_(gf) * cvec[tid] + sigmoidf_(gi) * tanhf(gg);
            float h  = sigmoidf_(go) * tanhf(c);
            cvec[tid] = c;
            v[HH + tid] = h;            // next step's h operand
            hsb[t * HH + tid] = h;      // encoder output
        }
        __syncthreads();
    }

    // ---- decoder temporal attention (h0=c0=0 => only W1[:,90:135] matters) --
    for (int idx = tid; idx < TT * DD; idx += NTHREADS) {
        const int t = idx / DD, j = idx % DD;
        float s = b1[j];
        for (int k = 0; k < HH; ++k) s += W1[j * (2 * HH + DD) + 2 * HH + k] * hsb[t * HH + k];
        dtmp[idx] = tanhf(s) * W2[j];
    }
    __syncthreads();
    // dscore[t] = b2 + sum_j dtmp[t][j]  (one wave per row, butterfly reduce)
    for (int trow = wave; trow < TT; trow += NTHREADS / 32) {
        float s = dtmp[trow * DD + lane];
        if (lane + 32 < DD) s += dtmp[trow * DD + lane + 32];
        s = waveSum(s);
        if (lane == 0) dscore[trow] = s + b2[0];
    }
    __syncthreads();
    // softmax over T=10, wave 0
    if (wave == 0) {
        float s0 = (lane < TT) ? dscore[lane] : -1e30f;
        float m  = waveMax(s0);
        float e0 = (lane < TT) ? __expf(s0 - m) : 0.0f;
        float inv = 1.0f / waveSum(e0);
        if (lane < TT) dscore[lane] = e0 * inv;
    }
    __syncthreads();
    if (tid < HH) {           // context = attn . hs
        float s = 0.0f;
#pragma unroll
        for (int t = 0; t < TT; ++t) s += dscore[t] * hsb[t * HH + tid];
        ctx[tid] = s;
    }
    __syncthreads();
    // decoder LSTM step with h0=c0=0: Whh term and f*c term vanish
    if (tid < NG) {
        float s = bih_d[tid] + bhh_d[tid];
        for (int k = 0; k < HH; ++k) s += Wih_d[tid * DD + k] * ctx[k];
        dg[tid] = s;
    }
    __syncthreads();
    if (tid < HH) {
        float gi = sigmoidf_(dg[tid]);
        float gg = tanhf(dg[2 * HH + tid]);
        float go = sigmoidf_(dg[3 * HH + tid]);
        float c  = gi * gg;
        h1v[tid] = go * tanhf(c);
    }
    __syncthreads();
    // y = [h1, context] @ Wfc^T + bfc
    if (tid < 45) {
        float s = bfc[tid];
        for (int j = 0; j < HH; ++j)
            s += Wfc[tid * (2 * HH) + j] * h1v[j] + Wfc[tid * (2 * HH) + HH + j] * ctx[j];
        out[tid] = s;
    }
}

extern "C" void kernel_launch(void* const* d_in, const int* in_sizes, int n_in,
                              void* d_out, int out_size, void* d_ws, size_t ws_size,
                              hipStream_t stream) {
    (void)in_sizes; (void)n_in; (void)out_size; (void)d_ws; (void)ws_size;
    const float* price = (const float*)d_in[0];
    const float* Wa    = (const float*)d_in[1];
    const float* ba    = (const float*)d_in[2];
    const float* Wih_e = (const float*)d_in[3];
    const float* Whh_e = (const float*)d_in[4];
    const float* bih_e = (const float*)d_in[5];
    const float* bhh_e = (const float*)d_in[6];
    const float* W1    = (const float*)d_in[7];
    const float* b1    = (const float*)d_in[8];
    const float* W2    = (const float*)d_in[9];
    const float* b2    = (const float*)d_in[10];
    const float* Wih_d = (const float*)d_in[11];
    // d_in[12] = Whh_d: unused (decoder h0 == 0)
    const float* bih_d = (const float*)d_in[13];
    const float* bhh_d = (const float*)d_in[14];
    const float* Wfc   = (const float*)d_in[15];
    const float* bfc   = (const float*)d_in[16];
    float* out = (float*)d_out;

    const size_t smemFloats =
        (size_t)MP * KP + TT * DD + TT * HH + NG + KP + MP + HH + DD + 64 +
        TT * DD + 16 + HH + NG + HH;
    const size_t smemBytes = smemFloats * sizeof(float);   // ~80 KB (WGP has 320 KB)

    // opt in to >64KB dynamic LDS (deterministic, called every launch)
    (void)hipFuncSetAttribute((const void*)lstm_price_fused,
                              hipFuncAttributeMaxDynamicSharedMemorySize,
                              (int)smemBytes);

    lstm_price_fused<<<dim3(1), dim3(NTHREADS), smemBytes, stream>>>(
        price, Wa, ba, Wih_e, Whh_e, bih_e, bhh_e,
        W1, b1, W2, b2, Wih_d, bih_d, bhh_d, Wfc, bfc, out);
}